// GAT_58617713656413
// MI455X (gfx1250) — compile-verified
//
#include <hip/hip_runtime.h>

typedef __attribute__((ext_vector_type(2))) float v2f;
typedef __attribute__((ext_vector_type(8))) float v8f;

#define HC 128   // H*C
#define NH 4     // heads
#define CC 32    // channels per head

// ---------------- GEMM: Y[N,128] = X[N,128] @ W[128,128] via V_WMMA_F32_16X16X4_F32
// 256 threads = 8 waves; each wave computes a 16x128 strip.
// W staged in LDS (64KB) in K-pair-interleaved layout so every B fragment
// {W[k][n], W[k+1][n]} is one contiguous ds_load_b64 into an even VGPR pair.
__global__ __launch_bounds__(256) void gemm128_wmma(const float* __restrict__ X,
                                                    const float* __restrict__ W,
                                                    float* __restrict__ Y, int Nn) {
    __shared__ float Wl[128 * 128];
    const int tid = threadIdx.x;
    // Wl[(k>>1)*256 + col*2 + (k&1)] = W[k*128 + col]
    for (int i = tid; i < 128 * 128; i += 256) {
        int k = i >> 7, col = i & 127;
        Wl[((k >> 1) << 8) + (col << 1) + (k & 1)] = W[i];
    }
    __syncthreads();

    const int wave = tid >> 5;
    const int lane = tid & 31;
    const int m    = lane & 15;          // row within tile (A), col within tile (B/D)
    const int kh   = (lane >> 4) << 1;   // K sub-pair: 0 or 2 (ISA 16x4 f32 A/B layout)
    const int m_base = blockIdx.x * 128 + wave * 16;

    int r = m_base + m; if (r >= Nn) r = Nn - 1;        // clamp (masked at store)
    const float* xrow = X + (size_t)r * 128;

    v8f acc[8];
#pragma unroll
    for (int t = 0; t < 8; ++t) acc[t] = (v8f){0,0,0,0,0,0,0,0};

    for (int kk = 0; kk < 128; kk += 4) {
        v2f a = *(const v2f*)(xrow + kk + kh);          // A[m][kk+kh .. +1]
        const float* bp = &Wl[(((kk + kh) >> 1) << 8) + (m << 1)];
#pragma unroll
        for (int t = 0; t < 8; ++t) {
            v2f b = *(const v2f*)(bp + t * 32);         // {W[k][n], W[k+1][n]}
            acc[t] = __builtin_amdgcn_wmma_f32_16x16x4_f32(
                false, a, false, b, (short)0, acc[t], false, false);
        }
    }

    const int hrow = (lane >> 4) * 8;                   // D: vgpr i -> row i + 8*(lane>>4)
#pragma unroll
    for (int t = 0; t < 8; ++t)
#pragma unroll
        for (int i = 0; i < 8; ++i) {
            int rr = m_base + hrow + i;
            if (rr < Nn) Y[(size_t)rr * 128 + t * 16 + m] = acc[t][i];
        }
}

// ---------------- attention coefficients: a_s[n,h] = <h[n,h,:], att_src[h,:]>
__global__ void attn_coef(const float* __restrict__ h, const float* __restrict__ att_s,
                          const float* __restrict__ att_d,
                          float* __restrict__ a_s, float* __restrict__ a_d, int total) {
    int idx = blockIdx.x * blockDim.x + threadIdx.x;
    if (idx >= total) return;
    int n = idx >> 2, hd = idx & 3;
    const float* hp = h + (size_t)n * HC + hd * CC;
    const float* as = att_s + hd * CC;
    const float* ad = att_d + hd * CC;
    float s = 0.f, d = 0.f;
#pragma unroll
    for (int c = 0; c < CC; ++c) { float v = hp[c]; s += v * as[c]; d += v * ad[c]; }
    a_s[idx] = s; a_d[idx] = d;
}

// ---------------- edge helpers
__device__ __forceinline__ void edge_sd(const int* __restrict__ ei, int e, int E,
                                        int& s, int& d) {
    if (e < E) { s = ei[e]; d = ei[E + e]; } else { s = e - E; d = s; }  // self-loops
}
// monotone order-preserving float<->u32 map (u32 atomicMax == float max)
__device__ __forceinline__ unsigned fenc(float f) {
    unsigned u = __float_as_uint(f);
    return (u & 0x80000000u) ? ~u : (u | 0x80000000u);
}
__device__ __forceinline__ float fdec(unsigned k) {
    return (k & 0x80000000u) ? __uint_as_float(k ^ 0x80000000u) : __uint_as_float(~k);
}
__device__ __forceinline__ float edge_e(const float* a_s, const float* a_d,
                                        int s, int d, int hd) {
    float v = a_s[s * NH + hd] + a_d[d * NH + hd];
    return v > 0.f ? v : 0.2f * v;                      // leaky_relu(0.2)
}

__global__ void edge_max(const int* __restrict__ ei, const float* __restrict__ a_s,
                         const float* __restrict__ a_d, unsigned* __restrict__ emax,
                         int E, int total) {
    int idx = blockIdx.x * blockDim.x + threadIdx.x;
    if (idx >= total) return;
    int e = idx >> 2, hd = idx & 3, s, d;
    edge_sd(ei, e, E, s, d);
    atomicMax(&emax[d * NH + hd], fenc(edge_e(a_s, a_d, s, d, hd)));
}

__global__ void edge_sum(const int* __restrict__ ei, const float* __restrict__ a_s,
                         const float* __restrict__ a_d, const unsigned* __restrict__ emax,
                         float* __restrict__ denom, int E, int total) {
    int idx = blockIdx.x * blockDim.x + threadIdx.x;
    if (idx >= total) return;
    int e = idx >> 2, hd = idx & 3, s, d;
    edge_sd(ei, e, E, s, d);
    float ex = __expf(edge_e(a_s, a_d, s, d, hd) - fdec(emax[d * NH + hd]));
    atomicAdd(&denom[d * NH + hd], ex);
}

// one thread per (edge, channel): wave32 covers 32 consecutive channels -> coalesced
__global__ void edge_agg(const int* __restrict__ ei, const float* __restrict__ a_s,
                         const float* __restrict__ a_d, const unsigned* __restrict__ emax,
                         const float* __restrict__ denom, const float* __restrict__ h,
                         float* __restrict__ outf, int E, int total) {
    int idx = blockIdx.x * blockDim.x + threadIdx.x;
    if (idx >= total) return;
    int e = idx >> 7, c = idx & 127, hd = c >> 5, s, d;
    edge_sd(ei, e, E, s, d);
    float ex = __expf(edge_e(a_s, a_d, s, d, hd) - fdec(emax[d * NH + hd]));
    float alpha = ex / (denom[d * NH + hd] + 1e-16f);
    atomicAdd(&outf[(size_t)d * HC + c], h[(size_t)s * HC + c] * alpha);
}

__global__ void bias_elu(float* __restrict__ feat, const float* __restrict__ b, int total) {
    int idx = blockIdx.x * blockDim.x + threadIdx.x;
    if (idx >= total) return;
    float v = feat[idx] + b[idx & 127];
    feat[idx] = v > 0.f ? v : expm1f(v);
}

__global__ void pool_sum(const float* __restrict__ feat, const int* __restrict__ batch,
                         float* __restrict__ pool, int total) {
    int idx = blockIdx.x * blockDim.x + threadIdx.x;
    if (idx >= total) return;
    atomicAdd(&pool[batch[idx >> 7] * HC + (idx & 127)], feat[idx]);
}
__global__ void pool_cnt(const int* __restrict__ batch, float* __restrict__ cnt, int Nn) {
    int n = blockIdx.x * blockDim.x + threadIdx.x;
    if (n >= Nn) return;
    atomicAdd(&cnt[batch[n]], 1.0f);
}

// pooled mean (+b2, folded here since pooling of a constant bias is the bias), FC, log_softmax
__global__ void final_head(const float* __restrict__ pool, const float* __restrict__ cnt,
                           const float* __restrict__ b2, const float* __restrict__ fcW,
                           const float* __restrict__ fcb, float* __restrict__ out, int Gn) {
    int g = threadIdx.x;
    if (g >= Gn) return;
    float inv = 1.0f / fmaxf(cnt[g], 1.0f);
    float logit[8];
#pragma unroll
    for (int k = 0; k < 8; ++k) logit[k] = fcb[k];
    for (int c = 0; c < HC; ++c) {
        float pv = pool[g * HC + c] * inv + b2[c];
#pragma unroll
        for (int k = 0; k < 8; ++k) logit[k] += pv * fcW[c * 8 + k];
    }
    float mx = logit[0];
#pragma unroll
    for (int k = 1; k < 8; ++k) mx = fmaxf(mx, logit[k]);
    float s = 0.f;
#pragma unroll
    for (int k = 0; k < 8; ++k) s += __expf(logit[k] - mx);
    float lse = mx + __logf(s);
#pragma unroll
    for (int k = 0; k < 8; ++k) out[g * 8 + k] = logit[k] - lse;
}

extern "C" void kernel_launch(void* const* d_in, const int* in_sizes, int n_in,
                              void* d_out, int out_size, void* d_ws, size_t ws_size,
                              hipStream_t stream) {
    const float* x    = (const float*)d_in[0];
    const int*   ei   = (const int*)d_in[1];
    const int*   batch= (const int*)d_in[2];
    const float* W1   = (const float*)d_in[3];
    const float* as1  = (const float*)d_in[4];
    const float* ad1  = (const float*)d_in[5];
    const float* b1   = (const float*)d_in[6];
    const float* W2   = (const float*)d_in[7];
    const float* as2  = (const float*)d_in[8];
    const float* ad2  = (const float*)d_in[9];
    const float* b2   = (const float*)d_in[10];
    const float* fcW  = (const float*)d_in[11];
    const float* fcb  = (const float*)d_in[12];
    float* out = (float*)d_out;

    const int Nn = in_sizes[2];          // 50000 (batch array length)
    const int E  = in_sizes[1] / 2;      // 1.6M
    const int EP = E + Nn;               // + self loops
    const int Gn = 64;

    // workspace carve-out
    char* ws = (char*)d_ws;
    size_t off = 0;
    auto carve = [&](size_t bytes) -> void* {
        void* p = ws + off;
        off = (off + bytes + 255) & ~(size_t)255;
        return p;
    };
    float*    hbuf  = (float*)carve((size_t)Nn * HC * 4);
    float*    feat  = (float*)carve((size_t)Nn * HC * 4);
    float*    a_s   = (float*)carve((size_t)Nn * NH * 4);
    float*    a_d   = (float*)carve((size_t)Nn * NH * 4);
    unsigned* emax  = (unsigned*)carve((size_t)Nn * NH * 4);
    float*    denom = (float*)carve((size_t)Nn * NH * 4);
    float*    pool  = (float*)carve((size_t)Gn * HC * 4);
    float*    cnt   = (float*)carve((size_t)Gn * 4);
    (void)ws_size; (void)n_in; (void)out_size;

    const int gemmBlocks = (Nn + 127) / 128;
    const int tAttn = Nn * NH;
    const int tEdge = EP * NH;           // 6.6M
    const int tAggr = EP * HC;           // 211.2M (< 2^31)
    const int tFeat = Nn * HC;

    auto run_layer = [&](const float* in, const float* W, const float* atts,
                         const float* attd) {
        gemm128_wmma<<<gemmBlocks, 256, 0, stream>>>(in, W, hbuf, Nn);
        attn_coef<<<(tAttn + 255) / 256, 256, 0, stream>>>(hbuf, atts, attd, a_s, a_d, tAttn);
        hipMemsetAsync(emax, 0, (size_t)Nn * NH * 4, stream);   // 0 == encoded -inf
        hipMemsetAsync(denom, 0, (size_t)Nn * NH * 4, stream);
        hipMemsetAsync(feat, 0, (size_t)Nn * HC * 4, stream);   // after gemm consumed it
        edge_max<<<(tEdge + 255) / 256, 256, 0, stream>>>(ei, a_s, a_d, emax, E, tEdge);
        edge_sum<<<(tEdge + 255) / 256, 256, 0, stream>>>(ei, a_s, a_d, emax, denom, E, tEdge);
        edge_agg<<<(tAggr + 255) / 256, 256, 0, stream>>>(ei, a_s, a_d, emax, denom,
                                                          hbuf, feat, E, tAggr);
    };

    // layer 1: feat = ELU(GATConv(x; W1) + b1)
    run_layer(x, W1, as1, ad1);
    bias_elu<<<(tFeat + 255) / 256, 256, 0, stream>>>(feat, b1, tFeat);

    // layer 2: feat = GATConv(feat; W2)   (b2 folded into final head)
    run_layer(feat, W2, as2, ad2);

    // global mean pool + FC + log_softmax
    hipMemsetAsync(pool, 0, (size_t)Gn * HC * 4, stream);
    hipMemsetAsync(cnt, 0, (size_t)Gn * 4, stream);
    pool_sum<<<(tFeat + 255) / 256, 256, 0, stream>>>(feat, batch, pool, tFeat);
    pool_cnt<<<(Nn + 255) / 256, 256, 0, stream>>>(batch, cnt, Nn);
    final_head<<<1, 64, 0, stream>>>(pool, cnt, b2, fcW, fcb, out, Gn);
}